// SAGEConv_71871982731728
// MI455X (gfx1250) — compile-verified
//
#include <hip/hip_runtime.h>

typedef float v2f __attribute__((ext_vector_type(2)));
typedef float v8f __attribute__((ext_vector_type(8)));

#define DFEAT 128   // D_in == D_out == 128

// ---------------------------------------------------------------------------
// Zero the workspace (summed accumulator + degree array) each call.
// ---------------------------------------------------------------------------
__global__ void zero_f32(float* __restrict__ p, long long n) {
  long long i = (long long)blockIdx.x * blockDim.x + threadIdx.x;
  long long stride = (long long)gridDim.x * blockDim.x;
  for (; i < n; i += stride) p[i] = 0.0f;
}

// ---------------------------------------------------------------------------
// Edge-parallel gather + scatter-add. One wave32 per edge: each lane loads a
// float4 (16B) of the 512B source feature row (fully coalesced b128 loads)
// and issues 4 global_atomic_add_f32 into the destination row. Degree counted
// by lane 0. All atomics land in L2 (dest array = 51MB << 192MB L2).
// ---------------------------------------------------------------------------
__global__ void sage_scatter(const float* __restrict__ feat,
                             const int* __restrict__ src,
                             const int* __restrict__ dst,
                             float* __restrict__ summed,
                             float* __restrict__ deg,
                             int E) {
  int gid  = blockIdx.x * blockDim.x + threadIdx.x;
  int edge = gid >> 5;
  int lane = threadIdx.x & 31;
  if (edge >= E) return;
  int s = __builtin_amdgcn_readfirstlane(src[edge]);
  int d = __builtin_amdgcn_readfirstlane(dst[edge]);
  if (lane == 0) atomicAdd(&deg[d], 1.0f);
  const float4 v = ((const float4*)(feat + (long long)s * DFEAT))[lane];
  float* drow = summed + (long long)d * DFEAT + lane * 4;
  atomicAdd(drow + 0, v.x);
  atomicAdd(drow + 1, v.y);
  atomicAdd(drow + 2, v.z);
  atomicAdd(drow + 3, v.w);
}

// ---------------------------------------------------------------------------
// Fused: out = feat @ W_self + (summed/max(deg,1)) @ W_neigh + b_self + b_neigh
// using V_WMMA_F32_16X16X4_F32 (exact fp32 tensor path, wave32).
// One wave per 16-node tile, 8 accumulator tiles covering all 128 out cols.
//
// Fragment layouts (ISA 7.12.2, wave32):
//   A 16x4 f32:  lane holds A[M=lane&15][kbase + (lane>>4)*2 + {0,1}]  (v2f)
//   B 4x16 f32:  lane holds B[kbase + (lane>>4)*2 + {0,1}][N=lane&15]  (v2f)
//   C/D 16x16:   vgpr v, lanes 0-15 -> (M=v, N=lane); lanes 16-31 -> (M=v+8)
// ---------------------------------------------------------------------------
__global__ __launch_bounds__(256)
void sage_gemm_wmma(const float* __restrict__ feat,
                    const float* __restrict__ summed,
                    const float* __restrict__ deg,
                    const float* __restrict__ Wself,
                    const float* __restrict__ bself,
                    const float* __restrict__ Wneigh,
                    const float* __restrict__ bneigh,
                    float* __restrict__ out,
                    int ntiles) {
  const int lane = threadIdx.x & 31;
  const int wave = threadIdx.x >> 5;
  const int tile = blockIdx.x * (blockDim.x >> 5) + wave;
  if (tile >= ntiles) return;   // wave-uniform: EXEC all-1s inside WMMA region

  const int mloc = lane & 15;          // M row (A) / N col (B,C) within tile
  const int kgrp = (lane >> 4) * 2;    // K sub-offset for A/B fragments
  const int row  = tile * 16 + mloc;

  const float invd = 1.0f / fmaxf(deg[row], 1.0f);

  // Accumulators pre-loaded with the (column-only) bias sum.
  v8f acc[8];
#pragma unroll
  for (int nt = 0; nt < 8; ++nt) {
    const int col = nt * 16 + mloc;
    const float bv = bself[col] + bneigh[col];
    acc[nt] = (v8f){bv, bv, bv, bv, bv, bv, bv, bv};
  }

  const float* fRow = feat   + (long long)row * DFEAT;
  const float* sRow = summed + (long long)row * DFEAT;

  for (int k = 0; k < DFEAT; k += 4) {
    const int kk = k + kgrp;
    v2f aS = *(const v2f*)(fRow + kk);          // self A-fragment
    v2f aN;                                     // mean-neighbor A-fragment
    aN.x = sRow[kk]     * invd;
    aN.y = sRow[kk + 1] * invd;

    // 8 independent self-WMMAs, then 8 independent neigh-WMMAs: consecutive
    // reuses of the same accumulator are 8 instructions apart (hides RAW).
#pragma unroll
    for (int nt = 0; nt < 8; ++nt) {
      const int col = nt * 16 + mloc;
      v2f bS;
      bS.x = Wself[(kk)     * DFEAT + col];
      bS.y = Wself[(kk + 1) * DFEAT + col];
      acc[nt] = __builtin_amdgcn_wmma_f32_16x16x4_f32(
          false, aS, false, bS, (short)0, acc[nt], false, false);
    }
#pragma unroll
    for (int nt = 0; nt < 8; ++nt) {
      const int col = nt * 16 + mloc;
      v2f bN;
      bN.x = Wneigh[(kk)     * DFEAT + col];
      bN.y = Wneigh[(kk + 1) * DFEAT + col];
      acc[nt] = __builtin_amdgcn_wmma_f32_16x16x4_f32(
          false, aN, false, bN, (short)0, acc[nt], false, false);
    }
  }

  // Write back per C/D layout: vgpr v -> row v (lanes 0-15) / v+8 (lanes 16-31)
  const int rbase = tile * 16 + (lane >> 4) * 8;
#pragma unroll
  for (int nt = 0; nt < 8; ++nt) {
    const int col = nt * 16 + mloc;
#pragma unroll
    for (int v = 0; v < 8; ++v) {
      out[(long long)(rbase + v) * DFEAT + col] = acc[nt][v];
    }
  }
}

// ---------------------------------------------------------------------------
extern "C" void kernel_launch(void* const* d_in, const int* in_sizes, int n_in,
                              void* d_out, int out_size, void* d_ws, size_t ws_size,
                              hipStream_t stream) {
  const float* feat   = (const float*)d_in[0];
  const int*   src    = (const int*)d_in[1];
  const int*   dst    = (const int*)d_in[2];
  const float* Wself  = (const float*)d_in[3];
  const float* bself  = (const float*)d_in[4];
  const float* Wneigh = (const float*)d_in[5];
  const float* bneigh = (const float*)d_in[6];
  float* out = (float*)d_out;

  const int N = in_sizes[0] / DFEAT;   // 100000
  const int E = in_sizes[1];           // 3200000

  // Workspace layout: [summed: N*128 f32][deg: N f32]  (~51.6 MB)
  float* summed = (float*)d_ws;
  float* deg    = summed + (long long)N * DFEAT;

  // 1) zero accumulators (contiguous region)
  const long long zn = (long long)N * DFEAT + N;
  zero_f32<<<1024, 256, 0, stream>>>(summed, zn);

  // 2) edge scatter: one wave32 per edge
  {
    const long long threads = (long long)E * 32;
    const int blk = 256;
    const long long blocks = (threads + blk - 1) / blk;
    sage_scatter<<<dim3((unsigned)blocks), blk, 0, stream>>>(
        feat, src, dst, summed, deg, E);
  }

  // 3) fused normalize + dual GEMM + bias via f32 WMMA
  {
    const int ntiles = N / 16;              // 6250 (exact)
    const int wavesPerBlock = 8;            // 256 threads
    const int blocks = (ntiles + wavesPerBlock - 1) / wavesPerBlock;
    sage_gemm_wmma<<<blocks, 256, 0, stream>>>(
        feat, summed, deg, Wself, bself, Wneigh, bneigh, out, ntiles);
  }
}